// SinkhornDistance_68736656605801
// MI455X (gfx1250) — compile-verified
//
#include <hip/hip_runtime.h>
#include <math.h>

// Problem constants (from reference: x,y = [8, 2048, 64] f32)
constexpr int   B        = 8;
constexpr int   P        = 2048;
constexpr int   D        = 64;
constexpr float EPS      = 0.1f;
constexpr float INV_EPS  = 10.0f;
constexpr int   MAX_ITER = 100;
constexpr float THRESH   = 0.1f;
constexpr int   CHUNK    = 8192;   // elements of pi/cost per workgroup

typedef __attribute__((ext_vector_type(2))) float v2f;
typedef __attribute__((ext_vector_type(8))) float v8f;

// ---------------------------------------------------------------------------
// Squared row norms of x and y: one wave32 per row (D=64 -> 2 floats/lane).
// ---------------------------------------------------------------------------
__global__ void norms_kernel(const float* __restrict__ x, const float* __restrict__ y,
                             float* __restrict__ xn, float* __restrict__ yn) {
    const int wave = threadIdx.x >> 5;
    const int lane = threadIdx.x & 31;
    const int row  = blockIdx.x * 8 + wave;           // 0 .. B*P-1
    const float* src = (blockIdx.y == 0) ? x : y;
    float*       dst = (blockIdx.y == 0) ? xn : yn;
    const float* p = src + (size_t)row * D + lane;
    const float a0 = p[0], a1 = p[32];
    float s = a0 * a0 + a1 * a1;
#pragma unroll
    for (int m = 16; m >= 1; m >>= 1) s += __shfl_xor(s, m, 32);
    if (lane == 0) dst[row] = s;
}

// ---------------------------------------------------------------------------
// C[b,i,j] = xn[i] + yn[j] - 2 * (x yT)[i,j] via V_WMMA_F32_16X16X4_F32.
// One wave computes one 16x16 tile; 8 waves/WG cover 16x128 of C.
// A (16x4 f32): lanes 0-15 hold M=0..15 with K={k0,k0+1}; lanes 16-31 K={k0+2,k0+3}.
// B (4x16 f32): symmetric (rows of y are columns of B).
// C/D (16x16 f32, 8 VGPRs): vgpr r -> M = r + 8*(lane>=16), N = lane&15.
// ---------------------------------------------------------------------------
__global__ void cost_wmma_kernel(const float* __restrict__ x, const float* __restrict__ y,
                                 const float* __restrict__ xn, const float* __restrict__ yn,
                                 float* __restrict__ Cm) {
    const int wave = threadIdx.x >> 5;
    const int lane = threadIdx.x & 31;
    const int b  = blockIdx.z;
    const int i0 = blockIdx.y * 16;
    const int j0 = blockIdx.x * 128 + wave * 16;
    const int m  = lane & 15;
    const int kh = (lane >> 4) << 1;                  // 0 or 2

    const float* xp = x + ((size_t)b * P + i0 + m) * D + kh;
    const float* yp = y + ((size_t)b * P + j0 + m) * D + kh;

    v8f acc = {};
#pragma unroll
    for (int k0 = 0; k0 < D; k0 += 4) {
        v2f a  = *(const v2f*)(xp + k0);
        v2f bb = *(const v2f*)(yp + k0);
        acc = __builtin_amdgcn_wmma_f32_16x16x4_f32(
            /*neg_a=*/false, a, /*neg_b=*/false, bb,
            /*c_mod=*/(short)0, acc, /*reuse_a=*/false, /*reuse_b=*/false);
    }

    const float ynj  = yn[(size_t)b * P + j0 + m];
    const int  mbase = (lane >> 4) << 3;              // 0 or 8
    float* cb = Cm + ((size_t)b * P + i0) * P + j0 + m;
#pragma unroll
    for (int r = 0; r < 8; ++r) {
        const float xni = xn[(size_t)b * P + i0 + mbase + r];
        cb[(size_t)(mbase + r) * P] = xni + ynj - 2.0f * acc[r];
    }
}

// ---------------------------------------------------------------------------
// Init: u = v = 0, convergence flag = 0.
// ---------------------------------------------------------------------------
__global__ void init_kernel(float* __restrict__ u, float* __restrict__ v, int* __restrict__ flag) {
    const int t = blockIdx.x * blockDim.x + threadIdx.x;
    if (t < B * P) { u[t] = 0.0f; v[t] = 0.0f; }
    if (t == 0) *flag = 0;
}

// ---------------------------------------------------------------------------
// Row logsumexp -> u update.  u cancels inside its own lse:
//   u_new = eps*log_mu - eps * lse_j((v_j - C_ij)/eps)
// One WG per (b,i); 8 values/thread cached in registers (single C read).
// Also records |u_new - u| into du[] for the convergence test.
// ---------------------------------------------------------------------------
__global__ void row_update_u(const float* __restrict__ Cm, const float* __restrict__ v,
                             float* __restrict__ u, float* __restrict__ du,
                             const int* __restrict__ flag, float eps_log_mu) {
    if (*(volatile const int*)flag) return;           // frozen after convergence
    const int b = blockIdx.y;
    const int i = blockIdx.x;
    const int t = threadIdx.x;
    const float* row = Cm + ((size_t)b * P + i) * P;
    const float* vb  = v + (size_t)b * P;

    float vals[8];
    float m = -INFINITY;
#pragma unroll
    for (int s = 0; s < 8; ++s) {
        const int j = t + s * 256;
        const float tv = (vb[j] - row[j]) * INV_EPS;
        vals[s] = tv;
        m = fmaxf(m, tv);
    }
    __shared__ float red[256];
    red[t] = m; __syncthreads();
    for (int w = 128; w >= 1; w >>= 1) { if (t < w) red[t] = fmaxf(red[t], red[t + w]); __syncthreads(); }
    const float rmax = red[0]; __syncthreads();

    float s = 0.0f;
#pragma unroll
    for (int k = 0; k < 8; ++k) s += __expf(vals[k] - rmax);
    red[t] = s; __syncthreads();
    for (int w = 128; w >= 1; w >>= 1) { if (t < w) red[t] += red[t + w]; __syncthreads(); }

    if (t == 0) {
        const float lse = rmax + __logf(red[0]);
        const float un  = eps_log_mu - EPS * lse;
        const size_t idx = (size_t)b * P + i;
        du[idx] = fabsf(un - u[idx]);
        u[idx]  = un;
    }
}

// ---------------------------------------------------------------------------
// Column logsumexp -> v update (online max/scaled-sum).
// WG = 32 columns x 8 row-groups so C reads stay 128B-coalesced.
// ---------------------------------------------------------------------------
__global__ void col_update_v(const float* __restrict__ Cm, const float* __restrict__ u,
                             float* __restrict__ v, const int* __restrict__ flag,
                             float eps_log_nu) {
    if (*(volatile const int*)flag) return;
    const int b = blockIdx.y;
    const int l = threadIdx.x & 31;
    const int g = threadIdx.x >> 5;                   // 0..7
    const int j = blockIdx.x * 32 + l;
    const float* ub = u + (size_t)b * P;
    const float* cb = Cm + (size_t)b * P * P + j;

    float m = -INFINITY, s = 0.0f;
    for (int i = g; i < P; i += 8) {
        const float tv = (ub[i] - cb[(size_t)i * P]) * INV_EPS;
        const float mn = fmaxf(m, tv);
        s = s * __expf(m - mn) + __expf(tv - mn);
        m = mn;
    }
    __shared__ float sm[8][32];
    __shared__ float ss[8][32];
    sm[g][l] = m; ss[g][l] = s;
    __syncthreads();
    if (g == 0) {
        float M = sm[0][l], S = ss[0][l];
#pragma unroll
        for (int q = 1; q < 8; ++q) {
            const float mq = sm[q][l], sq = ss[q][l];
            const float mn = fmaxf(M, mq);
            S = S * __expf(M - mn) + sq * __expf(mq - mn);
            M = mn;
        }
        v[(size_t)b * P + j] = eps_log_nu - EPS * (M + __logf(S));
    }
}

// ---------------------------------------------------------------------------
// err = mean_b(sum_i |du|) = sum(du)/B; set flag when < THRESH.
// Fixed-order reduction -> deterministic across graph replays.
// ---------------------------------------------------------------------------
__global__ void check_kernel(const float* __restrict__ du, int* __restrict__ flag) {
    if (*(volatile const int*)flag) return;
    const int t = threadIdx.x;
    float s = 0.0f;
    for (int k = t; k < B * P; k += 256) s += du[k];
    __shared__ float red[256];
    red[t] = s; __syncthreads();
    for (int w = 128; w >= 1; w >>= 1) { if (t < w) red[t] += red[t + w]; __syncthreads(); }
    if (t == 0 && (red[0] / (float)B) < THRESH) *flag = 1;
}

// ---------------------------------------------------------------------------
// pi = exp((u_i + v_j - C)/eps); per-block partial of sum(pi*C) for cost.
// ---------------------------------------------------------------------------
__global__ void pi_cost_kernel(const float* __restrict__ Cm, const float* __restrict__ u,
                               const float* __restrict__ v, float* __restrict__ pi,
                               float* __restrict__ partial) {
    const int b = blockIdx.y;
    const size_t batch0 = (size_t)b * P * P;
    const size_t base   = batch0 + (size_t)blockIdx.x * CHUNK;
    const float* ub = u + (size_t)b * P;
    const float* vb = v + (size_t)b * P;
    float acc = 0.0f;
#pragma unroll
    for (int s = 0; s < CHUNK / 256; ++s) {
        const size_t e  = base + threadIdx.x + (size_t)s * 256;
        const size_t ib = e - batch0;
        const int i = (int)(ib >> 11);
        const int j = (int)(ib & (size_t)(P - 1));
        const float c = Cm[e];
        const float p = __expf((ub[i] + vb[j] - c) * INV_EPS);
        pi[e] = p;
        acc += p * c;
    }
    __shared__ float red[256];
    red[threadIdx.x] = acc; __syncthreads();
    for (int w = 128; w >= 1; w >>= 1) { if (threadIdx.x < w) red[threadIdx.x] += red[threadIdx.x + w]; __syncthreads(); }
    if (threadIdx.x == 0) partial[(size_t)b * gridDim.x + blockIdx.x] = red[0];
}

__global__ void cost_reduce_kernel(const float* __restrict__ partial, float* __restrict__ cost,
                                   int nparts) {
    const int b = blockIdx.x;
    const int t = threadIdx.x;
    float s = 0.0f;
    for (int k = t; k < nparts; k += 256) s += partial[(size_t)b * nparts + k];
    __shared__ float red[256];
    red[t] = s; __syncthreads();
    for (int w = 128; w >= 1; w >>= 1) { if (t < w) red[t] += red[t + w]; __syncthreads(); }
    if (t == 0) cost[b] = red[0];
}

// ---------------------------------------------------------------------------
// Host launcher. d_out = [cost(8) | pi(B*P*P) | C(B*P*P)] f32.
// ---------------------------------------------------------------------------
extern "C" void kernel_launch(void* const* d_in, const int* in_sizes, int n_in,
                              void* d_out, int out_size, void* d_ws, size_t ws_size,
                              hipStream_t stream) {
    const float* x = (const float*)d_in[0];
    const float* y = (const float*)d_in[1];

    float* out  = (float*)d_out;
    float* cost = out;                                  // [B]
    float* pi   = out + B;                              // [B*P*P]
    float* Cm   = out + B + (size_t)B * P * P;          // [B*P*P]

    float* ws      = (float*)d_ws;
    float* u       = ws;                                // B*P
    float* v       = ws + 1 * B * P;                    // B*P
    float* du      = ws + 2 * B * P;                    // B*P
    float* xn      = ws + 3 * B * P;                    // B*P
    float* yn      = ws + 4 * B * P;                    // B*P
    float* partial = ws + 5 * B * P;                    // B * (P*P/CHUNK)
    int*   flag    = (int*)(ws + 5 * B * P + B * (P * P / CHUNK));

    const float eps_log_mu = EPS * logf(1.0f / (float)P + 1e-8f);
    const float eps_log_nu = eps_log_mu;
    const int   nparts     = P * P / CHUNK;             // 512

    init_kernel<<<(B * P) / 256, 256, 0, stream>>>(u, v, flag);
    norms_kernel<<<dim3(B * P / 8, 2), 256, 0, stream>>>(x, y, xn, yn);
    cost_wmma_kernel<<<dim3(P / 128, P / 16, B), 256, 0, stream>>>(x, y, xn, yn, Cm);

    for (int it = 0; it < MAX_ITER; ++it) {
        row_update_u<<<dim3(P, B), 256, 0, stream>>>(Cm, v, u, du, flag, eps_log_mu);
        col_update_v<<<dim3(P / 32, B), 256, 0, stream>>>(Cm, u, v, flag, eps_log_nu);
        check_kernel<<<1, 256, 0, stream>>>(du, flag);
    }

    pi_cost_kernel<<<dim3(nparts, B), 256, 0, stream>>>(Cm, u, v, pi, partial);
    cost_reduce_kernel<<<B, 256, 0, stream>>>(partial, cost, nparts);
}